// SparseAdam_75153337745819
// MI455X (gfx1250) — compile-verified
//
#include <hip/hip_runtime.h>
#include <cstdint>

// SparseAdam embedding update for MI455X (gfx1250, wave32).
// Memory-bound: ~1.85 GB moved -> ~80us floor at 23.3 TB/s. No matrix ops apply.
// Layout: D=128 f32 == one wave32 row with one float4 (b128) per lane.
// CDNA5 paths: global_load_async_to_lds_b128 + s_wait_asynccnt (ASYNCcnt DMA),
// NT temporal hints, global_prefetch_b8, f32 global atomics.

typedef float v4f __attribute__((ext_vector_type(4)));

#define WAVE 32
#define BLOCK 256
#define WPB (BLOCK / WAVE)   // 8 waves per block

// ---------------------------------------------------------------------------
// Kernel 1: zero a float region (vectorized b128 NT stores, scalar tail).
// ---------------------------------------------------------------------------
__global__ void sa_zero_kernel(float* __restrict__ p, long n) {
  const long stride = (long)gridDim.x * blockDim.x;
  const long tid = (long)blockIdx.x * blockDim.x + threadIdx.x;
  const long n4 = n >> 2;
  v4f z = (v4f)0.0f;
  for (long e = tid; e < n4; e += stride)
    __builtin_nontemporal_store(z, ((v4f*)p) + e);
  for (long e = (n4 << 2) + tid; e < n; e += stride)
    p[e] = 0.0f;
}

// ---------------------------------------------------------------------------
// Kernel 2: scatter-accumulate. One wave per grad row.
//   gsum[idx[i], :] += grad[i, :]   (f32 global atomics, L2-resident target)
//   cnt[idx[i]]     += 1.0          (lane 0)
// ---------------------------------------------------------------------------
__global__ void sa_scatter_kernel(const int* __restrict__ idx,
                                  const float* __restrict__ grad,
                                  float* gsum,   // V*D, pre-zeroed (aliases emb_out)
                                  float* cnt,    // V,   pre-zeroed (aliases step_out)
                                  int N, int D) {
  const int lane   = threadIdx.x & (WAVE - 1);
  const int wave   = (int)(((long)blockIdx.x * blockDim.x + threadIdx.x) >> 5);
  const int nwaves = (int)(((long)gridDim.x * blockDim.x) >> 5);

  for (int i = wave; i < N; i += nwaves) {
    const int row = idx[i];
    const float* g = grad + (long)i * D;
    float* dst = gsum + (long)row * D;
    // Warm the accumulator line toward L2 so the RMW hits in-cache.
    __builtin_prefetch(dst + lane * 4, 1, 3);
    if (lane == 0) atomicAdd(cnt + row, 1.0f);
    for (int d = lane * 4; d < D; d += 4 * WAVE) {
      v4f gv = __builtin_nontemporal_load((const v4f*)(g + d));  // read-once stream
      atomicAdd(dst + d + 0, gv.x);
      atomicAdd(dst + d + 1, gv.y);
      atomicAdd(dst + d + 2, gv.z);
      atomicAdd(dst + d + 3, gv.w);
    }
  }
}

// ---------------------------------------------------------------------------
// Kernel 3: Adam update. One wave per table row (b128 per lane).
// Touched rows: gsum row is DMA'd to LDS (global_load_async_to_lds_b128)
// overlapping the emb/mem/pow b128 NT loads; s_wait_asynccnt 0 then ds read.
// Untouched rows (~59%): pure 3-stream copy, no gsum read, no Adam math.
// gsum aliases emb_o and cnt aliases step_o: each row is read then
// overwritten by the single wave that owns it (pointers kept non-restrict).
// ---------------------------------------------------------------------------
__global__ __launch_bounds__(BLOCK)
void sa_update_kernel(const float* __restrict__ emb,
                      const float* __restrict__ step,
                      const float* __restrict__ mem,
                      const float* __restrict__ pw,
                      const float* gsum,  // == emb_o
                      const float* cnt,   // == step_o
                      float* emb_o,
                      float* step_o,
                      float* __restrict__ mem_o,
                      float* __restrict__ pw_o,
                      int V, int D) {
  __shared__ v4f smem4[WPB * WAVE];   // 512 B per wave, 4 KB per block

  const int lane   = threadIdx.x & (WAVE - 1);
  const int wib    = threadIdx.x >> 5;                 // wave-in-block
  const int wave   = (int)(((long)blockIdx.x * blockDim.x + threadIdx.x) >> 5);
  const int nwaves = (int)(((long)gridDim.x * blockDim.x) >> 5);

  // LDS byte offset of this lane's 16B staging slot (flat addr low 32 bits
  // == wave-relative LDS offset on gfx1250).
  const unsigned laddr = (unsigned)(uintptr_t)(&smem4[wib * WAVE + lane]);

  // exact f32 images of the reference's python-double constants
  const float B1f = 0.9f, B2f = 0.999f;
  const float OMB1 = (float)(1.0 - 0.9);     // 0.1f after f32 rounding
  const float OMB2 = (float)(1.0 - 0.999);   // 0.001f
  const float LRf = 0.001f, EPSf = 1e-8f;
  const float LOG2_B1 = -0.15200309344504995f;   // log2(0.9)
  const float LOG2_B2 = -0.0014434551646785005f; // log2(0.999)

  for (int v = wave; v < V; v += nwaves) {
    const long rb = (long)v * D;
    const float c = cnt[v];
    const float s = step[v];

    if (c > 0.0f) {                       // wave-uniform branch (one row/wave)
      const float s_new = s + 1.0f;
      const float c1 = 1.0f - exp2f(s_new * LOG2_B1);
      const float c2 = 1.0f - exp2f(s_new * LOG2_B2);
      const float inv_cnt = 1.0f / c;     // c >= 1 here

      for (int d = lane * 4; d < D; d += 4 * WAVE) {
        // Kick off async DMA of the gsum chunk into LDS (ASYNCcnt path),
        // then issue the three register streams so the DMA overlaps them.
        const unsigned long long ga =
            (unsigned long long)(uintptr_t)(gsum + rb + d);
        asm volatile("global_load_async_to_lds_b128 %0, %1, off"
                     :: "v"(laddr), "v"(ga) : "memory");
        v4f e = __builtin_nontemporal_load((const v4f*)(emb + rb + d));
        v4f m = __builtin_nontemporal_load((const v4f*)(mem + rb + d));
        v4f p = __builtin_nontemporal_load((const v4f*)(pw  + rb + d));
        asm volatile("s_wait_asynccnt 0" ::: "memory");
        v4f gs = smem4[wib * WAVE + lane];          // ds_load_b128

        v4f m_new, p_new, e_new;
#pragma unroll
        for (int k = 0; k < 4; ++k) {
          const float ga_k = gs[k] * inv_cnt;
          const float mn = B1f * m[k] + OMB1 * ga_k;
          const float pn = B2f * p[k] + OMB2 * ga_k * ga_k;
          const float mem_corr = mn / c1;
          const float pow_corr = pn / c2;
          const float upd = LRf * mem_corr / (sqrtf(pow_corr) + EPSf);
          m_new[k] = mn;
          p_new[k] = pn;
          e_new[k] = e[k] - upd;
        }
        // write-once streams: NT so 768MB of output doesn't thrash L2
        __builtin_nontemporal_store(m_new, (v4f*)(mem_o + rb + d));
        __builtin_nontemporal_store(p_new, (v4f*)(pw_o + rb + d));
        __builtin_nontemporal_store(e_new, (v4f*)(emb_o + rb + d)); // after gsum read
      }
      if (lane == 0) step_o[v] = s_new;   // overwrites cnt[v] (after the read)
    } else {
      // Untouched row: straight copy, skip gsum entirely.
      for (int d = lane * 4; d < D; d += 4 * WAVE) {
        v4f e = __builtin_nontemporal_load((const v4f*)(emb + rb + d));
        v4f m = __builtin_nontemporal_load((const v4f*)(mem + rb + d));
        v4f p = __builtin_nontemporal_load((const v4f*)(pw  + rb + d));
        __builtin_nontemporal_store(e, (v4f*)(emb_o + rb + d));
        __builtin_nontemporal_store(m, (v4f*)(mem_o + rb + d));
        __builtin_nontemporal_store(p, (v4f*)(pw_o  + rb + d));
      }
      if (lane == 0) step_o[v] = s;
    }
  }
}

// ---------------------------------------------------------------------------
// Launch: inputs (idx, grad, emb, state_step, state_mem, state_power);
// d_out = [emb_new (V*D) | step_new (V) | mem_new (V*D) | pow_new (V*D)].
// emb_new/step_new regions double as gsum/cnt scratch -> no d_ws needed.
// ---------------------------------------------------------------------------
extern "C" void kernel_launch(void* const* d_in, const int* in_sizes, int n_in,
                              void* d_out, int out_size, void* d_ws, size_t ws_size,
                              hipStream_t stream) {
  const int*   idx  = (const int*)d_in[0];
  const float* grad = (const float*)d_in[1];
  const float* emb  = (const float*)d_in[2];
  const float* step = (const float*)d_in[3];
  const float* mem  = (const float*)d_in[4];
  const float* pw   = (const float*)d_in[5];

  const int N = in_sizes[0];
  const int V = in_sizes[3];
  const int D = in_sizes[1] / N;   // 128

  float* out    = (float*)d_out;
  float* emb_o  = out;                     // V*D (scratch: gsum)
  float* step_o = out + (long)V * D;       // V   (scratch: cnt)
  float* mem_o  = step_o + V;              // V*D
  float* pw_o   = mem_o + (long)V * D;     // V*D

  // 1) zero gsum+cnt (contiguous region of V*D + V floats)
  const long zcount = (long)V * D + V;
  sa_zero_kernel<<<8192, BLOCK, 0, stream>>>(emb_o, zcount);

  // 2) scatter: one wave per grad row; 8 rows per 256-thread block
  const int blocks_sc = (N + WPB - 1) / WPB;
  sa_scatter_kernel<<<blocks_sc, BLOCK, 0, stream>>>(idx, grad, emb_o, step_o, N, D);

  // 3) update: one wave per table row
  const int blocks_up = (V + WPB - 1) / WPB;
  sa_update_kernel<<<blocks_up, BLOCK, 0, stream>>>(emb, step, mem, pw,
                                                    emb_o, step_o,
                                                    emb_o, step_o, mem_o, pw_o,
                                                    V, D);
}